// Spiking_8667244003799
// MI455X (gfx1250) — compile-verified
//
#include <hip/hip_runtime.h>

// ---------------------------------------------------------------------------
// Spiking conv block for MI455X (gfx1250, wave32).
//   0) one-time weight reorder [co][ci][kh][kw] -> [tap][co][ci]  (scratch)
//   1) conv1 (implicit GEMM, fp32 WMMA 16x16x4) + bias + clip(0, alpha)
//   2) sequential spiking scan over T (elementwise, in-place)
//   3) conv2 (implicit GEMM, fp32 WMMA 16x16x4) + bias
// fp32 WMMA chosen deliberately: the spike comparator (mem >= thr) amplifies
// low-precision conv error into spike flips, so f16/bf16 inputs are unsafe.
// ---------------------------------------------------------------------------

typedef __attribute__((ext_vector_type(2))) float v2f;
typedef __attribute__((ext_vector_type(8))) float v8f;

#define BB   32
#define TT   6
#define CCH  128            // channels (in == out)
#define HH   32
#define WW   32
#define NIMG (BB * TT)      // 192 independent images for the convs
#define XPITCH 130          // padded ci pitch of the input slab (bank spread)
#define WPITCH 132          // padded ci pitch of the weight tap (16B-aligned b128)
#define NTAPW  (9 * CCH * CCH)   // floats per reordered weight tensor

// ---- one-time: wgt[co][ci][kh][kw] -> wt[tap][co][ci], tap = kh*3+kw -------
__global__ __launch_bounds__(256)
void reorder_weights(const float* __restrict__ wsrc, float* __restrict__ wdst)
{
    int idx = blockIdx.x * 256 + threadIdx.x;      // over 9*128*128
    if (idx >= NTAPW) return;
    int tap = idx >> 14;                           // /16384
    int co  = (idx >> 7) & (CCH - 1);
    int ci  = idx & (CCH - 1);
    wdst[idx] = wsrc[(co * CCH + ci) * 9 + tap];   // coalesced writes
}

// One workgroup = 512 threads = 16 waves, handles (image n, output row h):
//   all 128 output channels x all 32 pixels of that row.
//   wave w: M-tile = w>>1 (co0 = 16*(w>>1)), N-tile = w&1 (w0 = 16*(w&1)).
// GEMM view per tap (kh,kw):  O[co, px] += W_tap[co, ci] * X[ci, h+kh-1, px+kw-1]
// K runs over ci in steps of 4 (V_WMMA_F32_16X16X4_F32), 9 taps x 32 steps.
template <bool CLIP>
__global__ __launch_bounds__(512, 1)
void conv3x3_wmma(const float* __restrict__ xin,    // [NIMG, C, H, W]
                  const float* __restrict__ wtrans, // [9, C(co), C(ci)] reordered
                  const float* __restrict__ bias,   // [C]
                  const float* __restrict__ alpha_p,
                  float* __restrict__ out)          // [NIMG, C, H, W]
{
    // xs declared FIRST so every B-fragment address fits the 16-bit DS
    // immediate-offset window (max ~53 KB).
    __shared__ float xs[3 * 34 * XPITCH];   // [r][col][ci pad 130]  53 040 B
    __shared__ float wsm[CCH * WPITCH];     // [co][ci pad 132]      67 584 B

    const int n    = blockIdx.x;            // image
    const int h    = blockIdx.y;            // output row
    const int tid  = threadIdx.x;
    const int wave = tid >> 5;
    const int lane = tid & 31;
    const int half = lane >> 4;             // 0: lanes 0-15, 1: lanes 16-31
    const int l    = lane & 15;
    const int co0  = (wave >> 1) * 16;      // M-tile base (output channel)
    const int w0   = (wave & 1) * 16;       // N-tile base (pixel column)

    // ---- stage transposed input slab: xs[r][col][ci] (rows h-1..h+1) ------
    // idx -> (r, ci, col) with col fastest => coalesced global reads;
    // LDS writes stride XPITCH apart => ~2-way banked, fine.
    const float* xn = xin + (size_t)n * (CCH * HH * WW);
    for (int idx = tid; idx < 3 * CCH * 34; idx += 512) {
        int r   = idx / (CCH * 34);
        int rem = idx - r * (CCH * 34);
        int ci  = rem / 34;
        int col = rem - ci * 34;
        int hr  = h - 1 + r;
        int wi  = col - 1;
        float v = 0.0f;
        if ((unsigned)hr < (unsigned)HH && (unsigned)wi < (unsigned)WW)
            v = xn[(ci * HH + hr) * WW + wi];
        xs[(r * 34 + col) * XPITCH + ci] = v;
    }

    v8f acc = {};   // 16x16 fp32 accumulator tile (8 VGPRs)

    for (int kh = 0; kh < 3; ++kh) {
        for (int kw = 0; kw < 3; ++kw) {
            const int tap = kh * 3 + kw;
            __syncthreads();   // protect wsm from previous tap's readers
            // ---- stage this tap's 128x128 weights via b128 copies ----
            const float* wt_tap = wtrans + (tap << 14);
            for (int idx = (tid << 2); idx < CCH * CCH; idx += 2048) {
                int co = idx >> 7;
                int ci = idx & (CCH - 1);
                float4 v = *(const float4*)(wt_tap + idx);     // coalesced b128
                *(float4*)&wsm[co * WPITCH + ci] = v;          // 16B aligned
            }
            __syncthreads();

            // prefetch next tap's 64KB weight block while we compute
            if (tap < 8)
                __builtin_prefetch(wt_tap + (CCH * CCH) + (tid << 5), 0, 3);

            // A row: this lane's output channel, K-half select (b64 pairs).
            const float* arow = &wsm[(co0 + l) * WPITCH + 2 * half];
            // B row: this wave's pixel (shifted by kw), ci contiguous.
            const float* brow = &xs[(kh * 34 + w0 + l + kw) * XPITCH + 2 * half];

            #pragma unroll 8
            for (int c = 0; c < CCH; c += 4) {
                // A 16x4: lane(l,half) -> A[M=l][K=2h], A[M=l][K=2h+1]
                v2f a = *(const v2f*)(arow + c);   // ds_load_b64
                // B 4x16: lane(l,half) -> B[K=2h][N=l], B[K=2h+1][N=l]
                v2f b = *(const v2f*)(brow + c);   // ds_load_b64
                acc = __builtin_amdgcn_wmma_f32_16x16x4_f32(
                        /*neg_a=*/false, a, /*neg_b=*/false, b,
                        /*c_mod=*/(short)0, acc,
                        /*reuse_a=*/false, /*reuse_b=*/false);
            }
        }
    }

    // ---- epilogue: bias (+ optional PACT clip), store row segment ----
    const float alpha = CLIP ? alpha_p[0] : 0.0f;
    float* orow = out + (size_t)n * (CCH * HH * WW) + h * WW + (w0 + l);
    #pragma unroll
    for (int g = 0; g < 8; ++g) {
        // D 16x16 layout: VGPR g, lanes 0-15 -> M=g ; lanes 16-31 -> M=g+8
        int   co = co0 + g + 8 * half;
        float v  = acc[g] + bias[co];
        if (CLIP) v = fminf(fmaxf(v, 0.0f), alpha);
        orow[(size_t)co * (HH * WW)] = v;
    }
}

// Sequential spiking neuron scan over T, one thread per (b, c, h, w) state.
// In-place: reads y[b,t,...], writes spike_train[b,t,...]*thr to same slot.
__global__ __launch_bounds__(256)
void spike_scan(float* __restrict__ y, const float* __restrict__ alpha_p)
{
    const int SPB = CCH * HH * WW;                    // states per batch elem
    size_t idx = (size_t)blockIdx.x * blockDim.x + threadIdx.x;
    const size_t S = (size_t)BB * SPB;                // 4 194 304 states
    if (idx >= S) return;

    const float thr = alpha_p[0];
    size_t b    = idx / SPB;
    size_t rest = idx - b * SPB;
    size_t base = b * (size_t)TT * SPB + rest;

    float mem = 0.5f * thr;
    float ss  = 0.0f;
    #pragma unroll
    for (int t = 0; t < TT; ++t) {
        size_t off = base + (size_t)t * SPB;
        float xt = y[off];
        mem += xt;
        float spike = (mem >= thr) ? 1.0f : 0.0f;
        mem -= thr * spike;
        ss  += spike;
        float inhib = ((mem <= -0.001f) && (ss > 0.0f)) ? 1.0f : 0.0f;
        mem += thr * inhib;
        ss  -= inhib;
        y[off] = (spike - inhib) * thr;               // in-place spike train
    }
}

extern "C" void kernel_launch(void* const* d_in, const int* in_sizes, int n_in,
                              void* d_out, int out_size, void* d_ws, size_t ws_size,
                              hipStream_t stream)
{
    const float* x      = (const float*)d_in[0];   // [B,T,C,H,W]
    const float* w_pre  = (const float*)d_in[1];   // [C,C,3,3]
    const float* b_pre  = (const float*)d_in[2];   // [C]
    const float* alpha  = (const float*)d_in[3];   // scalar
    const float* w_post = (const float*)d_in[4];   // [C,C,3,3]
    const float* b_post = (const float*)d_in[5];   // [C]
    float* out = (float*)d_out;                    // [B,T,C,H,W]

    // workspace layout: [wt_pre_t 576KB][wt_post_t 576KB][y 100.7MB]
    float* wt_pre_t  = (float*)d_ws;
    float* wt_post_t = wt_pre_t + NTAPW;
    float* y         = wt_post_t + NTAPW;

    // 0) reorder both weight tensors to [tap][co][ci]
    const int RB = (NTAPW + 255) / 256;
    reorder_weights<<<RB, 256, 0, stream>>>(w_pre,  wt_pre_t);
    reorder_weights<<<RB, 256, 0, stream>>>(w_post, wt_post_t);

    dim3 grid(NIMG, HH);                           // 192 images x 32 rows
    // 1) conv1 + bias + clip(0, alpha)  -> y
    conv3x3_wmma<true><<<grid, 512, 0, stream>>>(x, wt_pre_t, b_pre, alpha, y);
    // 2) spiking scan over T, in place on y
    const int S = BB * CCH * HH * WW;
    spike_scan<<<(S + 255) / 256, 256, 0, stream>>>(y, alpha);
    // 3) conv2 + bias -> out
    conv3x3_wmma<false><<<grid, 512, 0, stream>>>(y, wt_post_t, b_post, nullptr, out);
}